// MC_SS2D_12730283065734
// MI455X (gfx1250) — compile-verified
//
#include <hip/hip_runtime.h>

// ---------------- problem constants ----------------
constexpr int cB  = 2;
constexpr int cDM = 96;
constexpr int cH  = 64;
constexpr int cW  = 64;
constexpr int cL  = cH * cW;        // 4096
constexpr int cDI = 192;
constexpr int cN  = 16;
constexpr int cR  = 6;
constexpr int cK  = 8;
constexpr int cBL = cB * cL;        // 8192 rows (b,pixel)
constexpr int CDBL = cR + 2 * cN;   // 38 columns of x_dbl
constexpr int CH  = 64;             // scan chunk (steps per TDM tile)

typedef float v2f __attribute__((ext_vector_type(2)));
typedef float v8f __attribute__((ext_vector_type(8)));
typedef unsigned int v4u __attribute__((ext_vector_type(4)));
typedef int v4i __attribute__((ext_vector_type(4)));
typedef int v8i __attribute__((ext_vector_type(8)));

#define WMMA_F32X4(a, b, c) \
  __builtin_amdgcn_wmma_f32_16x16x4_f32(false, (a), false, (b), (short)0, (c), false, false)

__device__ __forceinline__ float siluf(float x) { return x / (1.0f + __expf(-x)); }
__device__ __forceinline__ float softplusf(float x) {
  return (x > 20.0f) ? x : log1pf(__expf(x));
}

// stream index s of direction k -> pixel raster index p (H=W=64)
__device__ __forceinline__ int map_pos(int k, int s) {
  int se = (k >= 4) ? (cL - 1 - s) : s;                 // flip directions
  if (k & 2) return ((se & 63) << 6) | (se >> 6);       // transposed raster
  return se;
}

// ---- Tensor Data Mover: 2D tile load Global -> LDS (D# per CDNA5 ISA ch.8) ----
// tile_d0 contiguous f32 along dim0, tile_d1 rows, row stride = stride_d0 elements.
__device__ __forceinline__ void tdm_load_2d_f32(unsigned lds_byte_addr,
                                                const float* gptr,
                                                unsigned tile_d0, unsigned tile_d1,
                                                unsigned tensor_d0, unsigned tensor_d1,
                                                unsigned stride_d0) {
  unsigned long long ga = (unsigned long long)(uintptr_t)gptr;
  v4u g0;
  g0.x = 1u;                                            // count=1, is_restore=0
  g0.y = lds_byte_addr;                                 // D#.lds_addr (bytes)
  g0.z = (unsigned)(ga & 0xFFFFFFFFu);                  // global_addr[31:0]
  g0.w = (unsigned)((ga >> 32) & 0x01FFFFFFu) | (2u << 30);  // addr[56:32] | type=2
  v8i g1;
  g1[0] = (int)(2u << 16);                              // data_size = 2 (4 bytes)
  g1[1] = (int)((tensor_d0 & 0xFFFFu) << 16);           // tensor_dim0[15:0]
  g1[2] = (int)(((tensor_d0 >> 16) & 0xFFFFu) | ((tensor_d1 & 0xFFFFu) << 16));
  g1[3] = (int)(((tensor_d1 >> 16) & 0xFFFFu) | ((tile_d0 & 0xFFFFu) << 16));
  g1[4] = (int)(tile_d1 & 0xFFFFu);                     // tile_dim1 (tile_dim2=0)
  g1[5] = (int)stride_d0;                               // tensor_dim0_stride[31:0]
  g1[6] = 0;
  g1[7] = 0;
  v4i g2 = {0, 0, 0, 0};
  v4i g3 = {0, 0, 0, 0};
  v8i g4 = {0, 0, 0, 0, 0, 0, 0, 0};
  __builtin_amdgcn_tensor_load_to_lds(g0, g1, g2, g3, g4, 0);
}

// ---------------- 1. per-pixel LayerNorm over DM, output column-major [2][DM][BL]
__global__ void k_ln(const float* __restrict__ f1, const float* __restrict__ f2,
                     const float* __restrict__ w1, const float* __restrict__ b1,
                     const float* __restrict__ w2, const float* __restrict__ b2,
                     float* __restrict__ fout) {
  int g = blockIdx.x * blockDim.x + threadIdx.x;
  if (g >= 2 * cBL) return;
  int mod = g / cBL, m = g % cBL;
  int b = m / cL, p = m % cL;
  const float* F = (mod ? f2 : f1) + (size_t)b * cDM * cL + p;
  const float* W = mod ? w2 : w1;
  const float* Bv = mod ? b2 : b1;
  float s = 0.f, s2 = 0.f;
  for (int c = 0; c < cDM; ++c) { float v = F[(size_t)c * cL]; s += v; s2 += v * v; }
  float mu = s / cDM;
  float var = s2 / cDM - mu * mu;
  float inv = rsqrtf(var + 1e-5f);
  float* O = fout + (size_t)mod * cDM * cBL + m;
  for (int c = 0; c < cDM; ++c)
    O[(size_t)c * cBL] = (F[(size_t)c * cL] - mu) * inv * W[c] + Bv[c];
}

// ---------------- 2. in_proj GEMM: [BL x 96] * [384 x 96]^T via f32 WMMA
__global__ void __launch_bounds__(256) k_inproj(const float* __restrict__ f,
                                                const float* __restrict__ w1,
                                                const float* __restrict__ w2,
                                                float* __restrict__ xpre,
                                                float* __restrict__ z) {
  const int mod = blockIdx.y;
  const float* F = f + (size_t)mod * cDM * cBL;
  const float* W = mod ? w2 : w1;
  float* XP = xpre + (size_t)mod * cBL * cDI;
  float* Z = z + (size_t)mod * cBL * cDI;
  int wave = threadIdx.x >> 5, lane = threadIdx.x & 31;
  int half = lane >> 4, lid = lane & 15;
  int tile = blockIdx.x * 8 + wave;
  const int NT = (2 * cDI) / 16;                 // 24 N-tiles
  int mt = tile / NT, nt = tile % NT;
  int m0 = mt * 16, n0 = nt * 16;
  const float* Wrow = W + (size_t)(n0 + lid) * cDM;
  v8f acc = {0, 0, 0, 0, 0, 0, 0, 0};
  for (int kb = 0; kb < cDM; kb += 4) {
    int ka = kb + 2 * half;
    v2f a, b;
    a.x = F[(size_t)ka * cBL + m0 + lid];
    a.y = F[(size_t)(ka + 1) * cBL + m0 + lid];
    b.x = Wrow[ka];
    b.y = Wrow[ka + 1];
    acc = WMMA_F32X4(a, b, acc);
  }
  int row0 = m0 + half * 8;
  int col = n0 + lid;
#pragma unroll
  for (int j = 0; j < 8; ++j) {
    float v = acc[j];
    int row = row0 + j;
    if (col < cDI) XP[(size_t)row * cDI + col] = v;
    else           Z[(size_t)row * cDI + (col - cDI)] = siluf(v);
  }
}

// ---------------- 3. depthwise 3x3 conv + SiLU : xpre[mod][m][d] -> xc[mod][m][d]
__global__ void k_dwconv(const float* __restrict__ xpre,
                         const float* __restrict__ cw1, const float* __restrict__ cb1,
                         const float* __restrict__ cw2, const float* __restrict__ cb2,
                         float* __restrict__ xc) {
  int idx = blockIdx.x * blockDim.x + threadIdx.x;
  if (idx >= 2 * cBL * cDI) return;
  int d = idx % cDI;
  int m = (idx / cDI) % cBL;
  int mod = idx / (cDI * cBL);
  const float* CW = mod ? cw2 : cw1;
  const float* CBv = mod ? cb2 : cb1;
  const float* XP = xpre + (size_t)mod * cBL * cDI;
  int b = m / cL, p = m % cL;
  int h = p >> 6, w = p & 63;
  float acc = CBv[d];
#pragma unroll
  for (int ky = -1; ky <= 1; ++ky) {
    int hh = h + ky;
    if (hh < 0 || hh >= cH) continue;
#pragma unroll
    for (int kx = -1; kx <= 1; ++kx) {
      int ww = w + kx;
      if (ww < 0 || ww >= cW) continue;
      int mp = b * cL + hh * cW + ww;
      acc += XP[(size_t)mp * cDI + d] * CW[d * 9 + (ky + 1) * 3 + (kx + 1)];
    }
  }
  xc[idx] = siluf(acc);
}

__global__ void k_zero(float* __restrict__ p, int n) {
  int i = blockIdx.x * blockDim.x + threadIdx.x;
  if (i < n) p[i] = 0.f;
}

// ---------------- 4. x_proj GEMM per stream k
__global__ void __launch_bounds__(256) k_xproj(const float* __restrict__ xc,
                                               const float* __restrict__ xw,
                                               float* __restrict__ xdbl) {
  const int k = blockIdx.y;
  const int mod = k & 1;
  int wave = threadIdx.x >> 5, lane = threadIdx.x & 31;
  int half = lane >> 4, lid = lane & 15;
  int tile = blockIdx.x * 8 + wave;              // (BL/16)=512 M-tiles x 3 N-tiles
  int mt = tile / 3, nt = tile % 3;
  int m0 = mt * 16, n0 = nt * 16;
  int bi = m0 / cL, sb = m0 % cL;
  int p = map_pos(k, sb + lid);
  const float* Arow = xc + ((size_t)mod * cBL + (size_t)bi * cL + p) * cDI;
  int c = n0 + lid;
  const float* Wrow = (c < CDBL) ? (xw + ((size_t)k * CDBL + c) * cDI) : (const float*)0;
  v8f acc = {0, 0, 0, 0, 0, 0, 0, 0};
  for (int kb = 0; kb < cDI; kb += 4) {
    int ka = kb + 2 * half;
    v2f a, b;
    a.x = Arow[ka];
    a.y = Arow[ka + 1];
    b.x = Wrow ? Wrow[ka] : 0.0f;
    b.y = Wrow ? Wrow[ka + 1] : 0.0f;
    acc = WMMA_F32X4(a, b, acc);
  }
  if (c < CDBL) {
    int s0 = sb + half * 8;
#pragma unroll
    for (int j = 0; j < 8; ++j)
      xdbl[((size_t)k * CDBL + c) * cBL + (size_t)bi * cL + s0 + j] = acc[j];
  }
}

// ---------------- 5. selective scan: one (b,k,d) stream per lane,
// x_dbl chunks DMA'd into double-buffered LDS by the Tensor Data Mover.
__global__ void __launch_bounds__(192) k_scan(const float* __restrict__ xdbl,
                                              const float* __restrict__ xc,
                                              const float* __restrict__ dtw,
                                              const float* __restrict__ dtb,
                                              const float* __restrict__ alog,
                                              const float* __restrict__ Dsp,
                                              float* __restrict__ yacc) {
  __shared__ float lds[2][CDBL * CH];   // [buf][c][s] : TDM tile row-major, 2 x 9728B
  const int bi = blockIdx.x / cK;
  const int k = blockIdx.x % cK;
  const int d = threadIdx.x;
  const int mod = k & 1;
  const int kd = k * cDI + d;
  float wdt[cR], An[cN], h[cN];
#pragma unroll
  for (int r = 0; r < cR; ++r) wdt[r] = dtw[(size_t)kd * cR + r];
#pragma unroll
  for (int n = 0; n < cN; ++n) {
    An[n] = -__expf(alog[(size_t)kd * cN + n]);
    h[n] = 0.f;
  }
  float bias = dtb[kd];
  float Dv = Dsp[kd];
  const float* XC = xc + ((size_t)mod * cBL + (size_t)bi * cL) * cDI;
  const float* gbase = xdbl + (size_t)k * CDBL * cBL + (size_t)bi * cL;  // + c*cBL + s
  const unsigned lds_base = (unsigned)(uintptr_t)(&lds[0][0]);
  const unsigned buf_bytes = (unsigned)(CDBL * CH * sizeof(float));
  const bool issuer = (threadIdx.x < 32);   // wave 0 drives the TDM

  if (issuer)   // prefetch chunk 0 into buffer 0
    tdm_load_2d_f32(lds_base, gbase, CH, CDBL, cBL, CDBL, cBL);

  const int NCH = cL / CH;
  for (int ci = 0; ci < NCH; ++ci) {
    int cur = ci & 1;
    if (issuer) __builtin_amdgcn_s_wait_tensorcnt(0);   // chunk ci landed in LDS
    __syncthreads();
    if (issuer && (ci + 1 < NCH))                        // overlap next DMA with math
      tdm_load_2d_f32(lds_base + (unsigned)(cur ^ 1) * buf_bytes,
                      gbase + (size_t)(ci + 1) * CH, CH, CDBL, cBL, CDBL, cBL);
    const float* buf = &lds[cur][0];
    const int ch = ci * CH;
    for (int s = 0; s < CH; ++s) {
      float x = bias;
#pragma unroll
      for (int r = 0; r < cR; ++r) x += buf[r * CH + s] * wdt[r];
      float delta = softplusf(x);
      int p = map_pos(k, ch + s);
      float u = XC[(size_t)p * cDI + d];
      float du = delta * u;
      float y = Dv * u;
#pragma unroll
      for (int n = 0; n < cN; ++n) {
        h[n] = __expf(delta * An[n]) * h[n] + du * buf[(cR + n) * CH + s];
        y += h[n] * buf[(cR + cN + n) * CH + s];
      }
      atomicAdd(&yacc[((size_t)bi * cL + p) * cDI + d], y);
    }
    __syncthreads();
  }
}

// ---------------- 6. out-LayerNorm over DI per pixel
__global__ void k_outln(const float* __restrict__ yacc, const float* __restrict__ w,
                        const float* __restrict__ b, float* __restrict__ yn) {
  int m = blockIdx.x * blockDim.x + threadIdx.x;
  if (m >= cBL) return;
  const float* Y = yacc + (size_t)m * cDI;
  float s = 0.f, s2 = 0.f;
  for (int d = 0; d < cDI; ++d) { float v = Y[d]; s += v; s2 += v * v; }
  float mu = s / cDI;
  float var = s2 / cDI - mu * mu;
  float inv = rsqrtf(var + 1e-5f);
  float* O = yn + (size_t)m * cDI;
  for (int d = 0; d < cDI; ++d) O[d] = (Y[d] - mu) * inv * w[d] + b[d];
}

// ---------------- 7. cc0: grouped conv (groups=DI, 2 in-ch per group)
__global__ void k_cc0(const float* __restrict__ yn, const float* __restrict__ z,
                      const float* __restrict__ w, const float* __restrict__ bb,
                      float* __restrict__ out) {
  int idx = blockIdx.x * blockDim.x + threadIdx.x;
  if (idx >= cBL * cDI) return;
  int g = idx % cDI, m = idx / cDI;
  int b = m / cL, p = m % cL;
  int h = p >> 6, ww = p & 63;
  int c0 = 2 * g, c1 = 2 * g + 1;
  int ch0 = (c0 < cDI) ? c0 : c0 - cDI;
  size_t zo0 = (c0 < cDI) ? 0 : (size_t)cBL * cDI;
  int ch1 = (c1 < cDI) ? c1 : c1 - cDI;
  size_t zo1 = (c1 < cDI) ? 0 : (size_t)cBL * cDI;
  float acc = bb[g];
#pragma unroll
  for (int ky = -1; ky <= 1; ++ky) {
    int hh = h + ky;
    if (hh < 0 || hh >= cH) continue;
#pragma unroll
    for (int kx = -1; kx <= 1; ++kx) {
      int wx = ww + kx;
      if (wx < 0 || wx >= cW) continue;
      int mp = b * cL + hh * cW + wx;
      float v0 = yn[(size_t)mp * cDI + ch0] * z[zo0 + (size_t)mp * cDI + ch0];
      float v1 = yn[(size_t)mp * cDI + ch1] * z[zo1 + (size_t)mp * cDI + ch1];
      acc += v0 * w[g * 18 + 0 * 9 + (ky + 1) * 3 + (kx + 1)];
      acc += v1 * w[g * 18 + 1 * 9 + (ky + 1) * 3 + (kx + 1)];
    }
  }
  out[idx] = siluf(acc);
}

// ---------------- 8. cc1: depthwise 3x3 + SiLU
__global__ void k_cc1(const float* __restrict__ in, const float* __restrict__ w,
                      const float* __restrict__ bb, float* __restrict__ out) {
  int idx = blockIdx.x * blockDim.x + threadIdx.x;
  if (idx >= cBL * cDI) return;
  int d = idx % cDI, m = idx / cDI;
  int b = m / cL, p = m % cL;
  int h = p >> 6, ww = p & 63;
  float acc = bb[d];
#pragma unroll
  for (int ky = -1; ky <= 1; ++ky) {
    int hh = h + ky;
    if (hh < 0 || hh >= cH) continue;
#pragma unroll
    for (int kx = -1; kx <= 1; ++kx) {
      int wx = ww + kx;
      if (wx < 0 || wx >= cW) continue;
      int mp = b * cL + hh * cW + wx;
      acc += in[(size_t)mp * cDI + d] * w[d * 9 + (ky + 1) * 3 + (kx + 1)];
    }
  }
  out[idx] = siluf(acc);
}

// ---------------- 9. cc2 1x1 conv (GEMM M=BL,N=DM,K=DI) + bias + residual
__global__ void __launch_bounds__(256) k_cc2(const float* __restrict__ a1,
                                             const float* __restrict__ w,
                                             const float* __restrict__ bb,
                                             const float* __restrict__ f1,
                                             const float* __restrict__ f2,
                                             float* __restrict__ out) {
  int wave = threadIdx.x >> 5, lane = threadIdx.x & 31;
  int half = lane >> 4, lid = lane & 15;
  int tile = blockIdx.x * 8 + wave;              // 512 M-tiles x 6 N-tiles
  int mt = tile / 6, nt = tile % 6;
  int m0 = mt * 16, n0 = nt * 16;
  const float* Arow = a1 + (size_t)(m0 + lid) * cDI;
  const float* Wrow = w + (size_t)(n0 + lid) * cDI;
  v8f acc = {0, 0, 0, 0, 0, 0, 0, 0};
  for (int kb = 0; kb < cDI; kb += 4) {
    int ka = kb + 2 * half;
    v2f a, b;
    a.x = Arow[ka];
    a.y = Arow[ka + 1];
    b.x = Wrow[ka];
    b.y = Wrow[ka + 1];
    acc = WMMA_F32X4(a, b, acc);
  }
  int col = n0 + lid;
  int row0 = m0 + half * 8;
#pragma unroll
  for (int j = 0; j < 8; ++j) {
    int row = row0 + j;
    int b_ = row / cL, p = row % cL;
    size_t o = (size_t)b_ * cDM * cL + (size_t)col * cL + p;
    out[o] = acc[j] + bb[col] + f1[o] + f2[o];
  }
}

// ---------------- host launcher ----------------
extern "C" void kernel_launch(void* const* d_in, const int* in_sizes, int n_in,
                              void* d_out, int out_size, void* d_ws, size_t ws_size,
                              hipStream_t stream) {
  (void)in_sizes; (void)n_in; (void)out_size; (void)ws_size;
  const float* feat1 = (const float*)d_in[0];
  const float* feat2 = (const float*)d_in[1];
  const float* ln1w = (const float*)d_in[2];
  const float* ln1b = (const float*)d_in[3];
  const float* ln2w = (const float*)d_in[4];
  const float* ln2b = (const float*)d_in[5];
  const float* ipw1 = (const float*)d_in[6];
  const float* ipw2 = (const float*)d_in[7];
  const float* cw1 = (const float*)d_in[8];
  const float* cb1 = (const float*)d_in[9];
  const float* cw2 = (const float*)d_in[10];
  const float* cb2 = (const float*)d_in[11];
  const float* xprojw = (const float*)d_in[12];
  const float* dtw = (const float*)d_in[13];
  const float* dtb = (const float*)d_in[14];
  const float* alogs = (const float*)d_in[15];
  const float* Ds = (const float*)d_in[16];
  const float* onw = (const float*)d_in[17];
  const float* onb = (const float*)d_in[18];
  const float* cc0w = (const float*)d_in[19];
  const float* cc0b = (const float*)d_in[20];
  const float* cc1w = (const float*)d_in[21];
  const float* cc1b = (const float*)d_in[22];
  const float* cc2w = (const float*)d_in[23];
  const float* cc2b = (const float*)d_in[24];

  float* ws = (float*)d_ws;
  float* f    = ws;                                   // [2][96][BL]   1,572,864 f32
  float* xpre = f + (size_t)2 * cDM * cBL;            // [2][BL][DI]   3,145,728
  float* z    = xpre + (size_t)2 * cBL * cDI;         // [2][BL][DI]   3,145,728
  float* xc   = z + (size_t)2 * cBL * cDI;            // [2][BL][DI]   3,145,728
  float* xdbl = xc + (size_t)2 * cBL * cDI;           // [K][38][BL]   2,490,368
  float* yacc = xdbl + (size_t)cK * CDBL * cBL;       // [BL][DI]      1,572,864
  // reuse regions whose producers are consumed by the time these are written:
  float* yn = f;                                      // [BL][DI]
  float* a0 = xpre;                                   // [BL][DI]
  float* a1 = xpre + (size_t)cBL * cDI;               // [BL][DI]
  float* outp = (float*)d_out;

  k_ln<<<(2 * cBL) / 256, 256, 0, stream>>>(feat1, feat2, ln1w, ln1b, ln2w, ln2b, f);
  k_inproj<<<dim3(((cBL / 16) * ((2 * cDI) / 16)) / 8, 2), 256, 0, stream>>>(f, ipw1, ipw2, xpre, z);
  k_dwconv<<<(2 * cBL * cDI) / 256, 256, 0, stream>>>(xpre, cw1, cb1, cw2, cb2, xc);
  k_zero<<<(cBL * cDI) / 256, 256, 0, stream>>>(yacc, cBL * cDI);
  k_xproj<<<dim3(((cBL / 16) * 3) / 8, cK), 256, 0, stream>>>(xc, xprojw, xdbl);
  k_scan<<<cB * cK, 192, 0, stream>>>(xdbl, xc, dtw, dtb, alogs, Ds, yacc);
  k_outln<<<cBL / 256, 256, 0, stream>>>(yacc, onw, onb, yn);
  k_cc0<<<(cBL * cDI) / 256, 256, 0, stream>>>(yn, z, cc0w, cc0b, a0);
  k_cc1<<<(cBL * cDI) / 256, 256, 0, stream>>>(a0, cc1w, cc1b, a1);
  k_cc2<<<((cBL / 16) * (cDM / 16)) / 8, 256, 0, stream>>>(a1, cc2w, cc2b, feat1, feat2, outp);
}